// RingKVCache_43645457662581
// MI455X (gfx1250) — compile-verified
//
#include <hip/hip_runtime.h>
#include <hip/hip_bf16.h>

// RingKVCache fused update for MI455X (gfx1250).
// Pure memory-bound op (~555 MB moved, 0 FLOPs) -> target is HBM roofline
// (~24us @ 23.3 TB/s). Strategy: single pass over each output tensor with
// 16B (B128) granularity; source select between cache and new val rows;
// bulk streaming through the CDNA5 async global<->LDS path (ASYNCcnt).

typedef float f4 __attribute__((ext_vector_type(4)));

// Exact parameter types of the gfx1250 async-LDS builtins (from hipcc diag):
//   arg0: __attribute__((vector_size(16))) int, address_space(1) pointer
//   arg1: same pointee, address_space(3) pointer
typedef int v4i_gcc __attribute__((vector_size(16)));
typedef __attribute__((address_space(1))) v4i_gcc* as1_v4i;
typedef __attribute__((address_space(3))) v4i_gcc* as3_v4i;

#define RKV_L     8192u
#define RKV_LM    8191u
#define RKV_S     512u
#define RKV_W     4096u   // window = L/2
#define RKV_ROW4  32u     // float4 per row of D=128
#define RKV_TPB   256u
#define RKV_CHUNK 8u      // float4 per thread per block

#define HAS_ASYNC_LDS (__has_builtin(__builtin_amdgcn_global_load_async_to_lds_b128) && \
                       __has_builtin(__builtin_amdgcn_global_store_async_from_lds_b128))

__device__ __forceinline__ void rkv_wait_async0() {
#if __has_builtin(__builtin_amdgcn_s_wait_asynccnt)
    __builtin_amdgcn_s_wait_asynccnt(0);
#else
    asm volatile("s_wait_asynccnt 0x0" ::: "memory");
#endif
}

// dst[b,h,l,:] = ( (l-start) mod L < S ) ? val[b,h,(l-start) mod L,:] : cache[b,h,l,:]
// One grid covers one tensor: 4096 blocks * 256 threads * 8 float4 = 8,388,608 float4.
__global__ __launch_bounds__(RKV_TPB) void ring_copy_scatter(
    const float* __restrict__ cache,
    const float* __restrict__ val,
    float* __restrict__ dst,
    const int* __restrict__ input_pos)
{
    const unsigned start = (unsigned)(*input_pos);          // wave-uniform s_load
    const unsigned tid   = threadIdx.x;
    const unsigned base  = blockIdx.x * (RKV_TPB * RKV_CHUNK);

#if HAS_ASYNC_LDS
    // Stream through LDS with CDNA5 async copies: 8 loads in flight per wave,
    // one ASYNCcnt drain, 8 stores, drain. 32KB LDS staging per block.
    __shared__ f4 stage[RKV_CHUNK][RKV_TPB];
    #pragma unroll
    for (unsigned i = 0; i < RKV_CHUNK; ++i) {
        const unsigned e  = base + i * RKV_TPB + tid;       // flat float4 index
        const unsigned d4 = e & (RKV_ROW4 - 1u);
        const unsigned l  = (e >> 5) & RKV_LM;
        const unsigned bh = e >> 18;                        // 32 (b,h) pairs
        const unsigned s  = (l - start) & RKV_LM;           // (l-start) mod L
        const f4* src = (s < RKV_S)
            ? ((const f4*)val)   + (bh * (RKV_S * RKV_ROW4) + s * RKV_ROW4 + d4)
            : ((const f4*)cache) + e;
        __builtin_amdgcn_global_load_async_to_lds_b128(
            (as1_v4i)src, (as3_v4i)&stage[i][tid], 0, 0);
    }
    rkv_wait_async0();
    #pragma unroll
    for (unsigned i = 0; i < RKV_CHUNK; ++i) {
        const unsigned e = base + i * RKV_TPB + tid;
        __builtin_amdgcn_global_store_async_from_lds_b128(
            (as1_v4i)(((f4*)dst) + e), (as3_v4i)&stage[i][tid], 0, 0);
    }
    rkv_wait_async0();
#else
    // Fallback: direct VGPR float4 copy (global_load_b128 / global_store_b128).
    #pragma unroll
    for (unsigned i = 0; i < RKV_CHUNK; ++i) {
        const unsigned e  = base + i * RKV_TPB + tid;
        const unsigned d4 = e & (RKV_ROW4 - 1u);
        const unsigned l  = (e >> 5) & RKV_LM;
        const unsigned bh = e >> 18;
        const unsigned s  = (l - start) & RKV_LM;
        const f4* src = (s < RKV_S)
            ? ((const f4*)val)   + (bh * (RKV_S * RKV_ROW4) + s * RKV_ROW4 + d4)
            : ((const f4*)cache) + e;
        ((f4*)dst)[e] = *src;
    }
#endif
}

// mask[q, l] = valid ? 0 : -inf, with cp[l] reconstructed analytically.
// 4096 blocks * 256 threads, each thread emits one float4 (4 consecutive l).
__global__ __launch_bounds__(RKV_TPB) void ring_mask(
    const int* __restrict__ cache_positions,
    const int* __restrict__ input_pos,
    float* __restrict__ mask)
{
    const int start = *input_pos;
    const unsigned idx = blockIdx.x * RKV_TPB + threadIdx.x;  // 0 .. S*L/4
    const unsigned q   = idx >> 11;                            // L/4 = 2048 per row
    const unsigned l0  = (idx & 2047u) << 2;
    const int pos_q = start + (int)q;

    f4 r;
    #pragma unroll
    for (unsigned j = 0; j < 4; ++j) {
        const unsigned l = l0 + j;
        const unsigned s = (l - (unsigned)start) & RKV_LM;
        int cp;
        if (s < RKV_S)                cp = start + (int)s;       // freshly scattered slot
        else if ((int)l < start)      cp = cache_positions[l];   // previously written slot
        else                          cp = -1;                   // never written
        const int  delta = pos_q - cp;
        const bool valid = (cp >= 0) & (delta >= 0) & (delta < (int)RKV_W);
        r[j] = valid ? 0.0f : -__builtin_inff();
    }
    ((f4*)mask)[idx] = r;
}

extern "C" void kernel_launch(void* const* d_in, const int* in_sizes, int n_in,
                              void* d_out, int out_size, void* d_ws, size_t ws_size,
                              hipStream_t stream) {
    (void)in_sizes; (void)n_in; (void)out_size; (void)d_ws; (void)ws_size;

    const float* k_cache        = (const float*)d_in[0];
    const float* v_cache        = (const float*)d_in[1];
    const float* k_val          = (const float*)d_in[2];
    const float* v_val          = (const float*)d_in[3];
    const int*   cache_positions= (const int*)d_in[4];
    const int*   input_pos      = (const int*)d_in[5];

    float* out   = (float*)d_out;
    const size_t kv_elems = 2ull * 16ull * 8192ull * 128ull;   // 33,554,432
    float* k_out = out;
    float* v_out = out + kv_elems;
    float* mask  = out + 2ull * kv_elems;

    // copy/scatter: (kv_elems/4) float4 per tensor / (256*8) per block = 4096 blocks
    const unsigned copy_blocks = (unsigned)((kv_elems / 4ull) / (RKV_TPB * RKV_CHUNK));
    ring_copy_scatter<<<copy_blocks, RKV_TPB, 0, stream>>>(k_cache, k_val, k_out, input_pos);
    ring_copy_scatter<<<copy_blocks, RKV_TPB, 0, stream>>>(v_cache, v_val, v_out, input_pos);

    // mask: S*L/4 float4 / 256 threads = 4096 blocks
    const unsigned mask_blocks = (RKV_S * RKV_L / 4u) / RKV_TPB;
    ring_mask<<<mask_blocks, RKV_TPB, 0, stream>>>(cache_positions, input_pos, mask);
}